// DeconvCG_17712445129211
// MI455X (gfx1250) — compile-verified
//
#include <hip/hip_runtime.h>

// ---------------------------------------------------------------------------
// CG deconvolution for MI455X (gfx1250).
//  - 11x11 convs run as banded-Toeplitz GEMMs on V_WMMA_F32_16X16X4_F32
//    (the only f32 WMMA shape; f16 would wreck CG convergence over 100 iters).
//  - A-fragments come from a zero-guard-padded LDS table -> branch-free
//    ds_load_2addr, no exec-mask juggling, one s_wait_dscnt per 7 WMMAs.
//  - 3x3 reg convs fused into one precomputed 5x5 kernel (exact in interior).
//  - All reductions are fixed-order trees (graph replay must be deterministic).
// ---------------------------------------------------------------------------

typedef __attribute__((ext_vector_type(2))) float v2f;
typedef __attribute__((ext_vector_type(8))) float v8f;

// Only the device pass declares amdgcn builtins; do not probe on host.
#if defined(__HIP_DEVICE_COMPILE__)
#if !__has_builtin(__builtin_amdgcn_wmma_f32_16x16x4_f32)
#error "device pass: missing __builtin_amdgcn_wmma_f32_16x16x4_f32"
#endif
#endif

#define CC   3
#define HH   512
#define WW   512
#define HWSZ (HH * WW)
#define CHW  (CC * HWSZ)
#define NITER 100
#define EWB  512   // blocks for elementwise kernels (also #partials)
#define EWT  256   // threads for elementwise kernels

// Padded Toeplitz row: wpad[dy][16+off] = w[dy][off] for off in [0,10], else 0.
// Lane reads index (16 + j0 - ml) with j0 in [0,27], ml in [0,15] -> [1,43].
#define WPAD 44

// ---------------------------------------------------------------------------
// 11x11 depthwise cross-correlation, SAME zero padding, via WMMA f32 16x16x4.
// One wave32 per 16x16 output tile; M = x-offset, N = y-offset in the tile.
// Per kernel row dy: out[m,n] += sum_j T_dy[m,j] * in[y0+n+dy-5, x0+j-5],
// T_dy[m,j] = w[dy, j-m] for 0<=j-m<=10 (banded Toeplitz), j in [0,28).
// ---------------------------------------------------------------------------
__global__ __launch_bounds__(256) void conv11_wmma(const float* __restrict__ src,
                                                   float* __restrict__ dst,
                                                   const float* __restrict__ w121) {
    __shared__ float wpad[11 * WPAD];    // zero-guarded Toeplitz rows
    __shared__ float slab[8][26 * 29];   // per-wave input slab, stride 29

    const int tid  = threadIdx.x;
    const int lane = tid & 31;
    const int wid  = tid >> 5;

    // Branch-free A-table: zeros outside the band, kernel row inside.
    for (int i = tid; i < 11 * WPAD; i += 256) {
        const int dy  = i / WPAD;
        const int off = i - dy * WPAD - 16;
        wpad[i] = (off >= 0 && off <= 10) ? w121[dy * 11 + off] : 0.0f;
    }

    const int tile = blockIdx.x * 8 + wid;           // 384 blocks * 8 waves = 3072 tiles
    const int c    = tile >> 10;                     // / (32*32)
    const int rem  = tile & 1023;
    const int y0   = (rem >> 5) << 4;
    const int x0   = (rem & 31) << 4;

    // Stage 26 rows x 28 cols (zero-padded at image borders) into LDS.
    const float* sc = src + c * HWSZ;
    for (int idx = lane; idx < 26 * 28; idx += 32) {
        const int sy = idx / 28;
        const int sx = idx - sy * 28;
        const int gy = y0 + sy - 5;
        const int gx = x0 + sx - 5;
        float v = 0.0f;
        if (gy >= 0 && gy < HH && gx >= 0 && gx < WW) v = sc[gy * WW + gx];
        slab[wid][sy * 29 + sx] = v;
    }
    __syncthreads();

    const int half = lane >> 4;    // K-half select for A/B fragments
    const int ml   = lane & 15;    // M for A, N for B/D

    v8f acc = {0.f, 0.f, 0.f, 0.f, 0.f, 0.f, 0.f, 0.f};

#if defined(__HIP_DEVICE_COMPILE__)
#pragma unroll
    for (int dy = 0; dy < 11; ++dy) {
        const float* ar = &wpad[dy * WPAD + 16 - ml];      // a = T_dy[ml, j0..j0+1]
        const float* br = &slab[wid][(ml + dy) * 29];      // b = in row (n+dy)
        v2f av[7], bv[7];
#pragma unroll
        for (int ch = 0; ch < 7; ++ch) {                   // stage 14 ds_load_2addr
            const int j0 = ch * 4 + half * 2;
            av[ch].x = ar[j0];
            av[ch].y = ar[j0 + 1];
            bv[ch].x = br[j0];
            bv[ch].y = br[j0 + 1];
        }
#pragma unroll
        for (int ch = 0; ch < 7; ++ch) {                   // 7 back-to-back WMMAs
            acc = __builtin_amdgcn_wmma_f32_16x16x4_f32(
                false, av[ch], false, bv[ch], (short)0, acc, false, false);
        }
    }
#endif

    // D layout: lane n holds M = v (+8 for lanes 16..31) -> 8 consecutive x's.
    float* dp = dst + c * HWSZ + (y0 + ml) * WW + x0 + half * 8;
    *(float4*)(dp)     = make_float4(acc[0], acc[1], acc[2], acc[3]);
    *(float4*)(dp + 4) = make_float4(acc[4], acc[5], acc[6], acc[7]);
}

// ---------------------------------------------------------------------------
// Deterministic 256-thread block tree reduction.
// ---------------------------------------------------------------------------
__device__ __forceinline__ float blockReduce256(float v, float* red) {
    const int tid = threadIdx.x;
    red[tid] = v;
    __syncthreads();
    for (int s = 128; s > 0; s >>= 1) {
        if (tid < s) red[tid] += red[tid + s];
        __syncthreads();
    }
    return red[0];
}

// Ap += R (*) p  (fused 5x5 regularizer, SAME zero pad), and partial p.Ap sums.
__global__ __launch_bounds__(256) void conv5_add_dot(const float* __restrict__ p,
                                                     float* __restrict__ Ap,
                                                     const float* __restrict__ R,
                                                     float* __restrict__ partials) {
    __shared__ float red[256];
    float local = 0.0f;
    for (int i = blockIdx.x * blockDim.x + threadIdx.x; i < CHW;
         i += gridDim.x * blockDim.x) {
        const int c    = i / HWSZ;
        const int rem2 = i - c * HWSZ;
        const int y    = rem2 >> 9;
        const int x    = rem2 & 511;
        const float* pc = p + c * HWSZ;
        float s = 0.0f;
#pragma unroll
        for (int dy = 0; dy < 5; ++dy) {
            const int yy = y + dy - 2;
            if (yy < 0 || yy >= HH) continue;
#pragma unroll
            for (int dx = 0; dx < 5; ++dx) {
                const int xx = x + dx - 2;
                if (xx < 0 || xx >= WW) continue;
                s += R[dy * 5 + dx] * pc[yy * WW + xx];
            }
        }
        const float apv = Ap[i] + s;
        Ap[i] = apv;
        local += p[i] * apv;
    }
    const float tot = blockReduce256(local, red);
    if (threadIdx.x == 0) partials[blockIdx.x] = tot;
}

// Single-block finalize: out[0] = sum(partials[0..511]) in a fixed order.
__global__ __launch_bounds__(256) void reduce512(const float* __restrict__ partials,
                                                 float* __restrict__ out) {
    __shared__ float red[256];
    const float v = partials[threadIdx.x] + partials[threadIdx.x + 256];
    const float tot = blockReduce256(v, red);
    if (threadIdx.x == 0) out[0] = tot;
}

// x = img; r = p = b - Ax; partial sums of r.r
__global__ __launch_bounds__(256) void init_state(const float* __restrict__ img,
                                                  const float* __restrict__ b,
                                                  const float* __restrict__ Ax,
                                                  float* __restrict__ x,
                                                  float* __restrict__ r,
                                                  float* __restrict__ p,
                                                  float* __restrict__ partials) {
    __shared__ float red[256];
    float local = 0.0f;
    for (int i = blockIdx.x * blockDim.x + threadIdx.x; i < CHW;
         i += gridDim.x * blockDim.x) {
        x[i] = img[i];
        const float rn = b[i] - Ax[i];
        r[i] = rn;
        p[i] = rn;
        local += rn * rn;
    }
    const float tot = blockReduce256(local, red);
    if (threadIdx.x == 0) partials[blockIdx.x] = tot;
}

// alpha = rr_old/pAp; x += alpha p; r -= alpha Ap; partial sums of r_new.r_new
__global__ __launch_bounds__(256) void axpy_update(float* __restrict__ x,
                                                   float* __restrict__ r,
                                                   const float* __restrict__ p,
                                                   const float* __restrict__ Ap,
                                                   const float* __restrict__ rr_old,
                                                   const float* __restrict__ pap,
                                                   float* __restrict__ partials) {
    __shared__ float red[256];
    const float alpha = rr_old[0] / pap[0];
    float local = 0.0f;
    for (int i = blockIdx.x * blockDim.x + threadIdx.x; i < CHW;
         i += gridDim.x * blockDim.x) {
        x[i] += alpha * p[i];
        const float rn = r[i] - alpha * Ap[i];
        r[i] = rn;
        local += rn * rn;
    }
    const float tot = blockReduce256(local, red);
    if (threadIdx.x == 0) partials[blockIdx.x] = tot;
}

// beta = rr_new/rr_old; p = r + beta p
__global__ __launch_bounds__(256) void p_update(float* __restrict__ p,
                                                const float* __restrict__ r,
                                                const float* __restrict__ rr_new,
                                                const float* __restrict__ rr_old) {
    const float beta = rr_new[0] / rr_old[0];
    for (int i = blockIdx.x * blockDim.x + threadIdx.x; i < CHW;
         i += gridDim.x * blockDim.x) {
        p[i] = r[i] + beta * p[i];
    }
}

// Build K, flip(K), fused 5x5 reg kernel R = sum_i w_i^2 * corr(G_i, G_i);
// zero the per-iteration scalar slots.
__global__ void prep(const float* __restrict__ kern, const float* __restrict__ rw,
                     const float* __restrict__ rks, float* __restrict__ Kw,
                     float* __restrict__ Kfw, float* __restrict__ R,
                     float* __restrict__ rr, float* __restrict__ pap) {
    const int t = threadIdx.x;
    if (t < 121) {
        Kw[t]  = kern[t];
        Kfw[t] = kern[120 - t];   // flip both dims
    }
    if (t < 128) { rr[t] = 0.0f; pap[t] = 0.0f; }
    if (t < 25) {
        const int dy = t / 5, dx = t - dy * 5;
        float s = 0.0f;
        for (int i = 0; i < 2; ++i) {
            const float w2 = rw[i] * rw[i];
            for (int uy = 0; uy < 3; ++uy)
                for (int ux = 0; ux < 3; ++ux) {
                    const int vy = uy + dy - 2, vx = ux + dx - 2;
                    if (vy >= 0 && vy < 3 && vx >= 0 && vx < 3)
                        s += w2 * rks[i * 9 + uy * 3 + ux] * rks[i * 9 + vy * 3 + vx];
                }
        }
        R[t] = s;
    }
}

extern "C" void kernel_launch(void* const* d_in, const int* in_sizes, int n_in,
                              void* d_out, int out_size, void* d_ws, size_t ws_size,
                              hipStream_t stream) {
    const float* img  = (const float*)d_in[0];   // [3,512,512]
    const float* kern = (const float*)d_in[1];   // [11,11]
    const float* rw   = (const float*)d_in[2];   // [2]
    const float* rks  = (const float*)d_in[3];   // [2,3,3]
    float* x  = (float*)d_out;                   // CG iterate lives in d_out
    float* ws = (float*)d_ws;

    float* t    = ws;                // conv intermediate
    float* Ap   = ws + 1 * (size_t)CHW;
    float* b    = ws + 2 * (size_t)CHW;
    float* r    = ws + 3 * (size_t)CHW;
    float* p    = ws + 4 * (size_t)CHW;
    float* misc = ws + 5 * (size_t)CHW;
    float* Kw  = misc;          // 128
    float* Kfw = misc + 128;    // 128
    float* R   = misc + 256;    // 32
    float* rr  = misc + 288;    // 128 scalar slots (rr[it])
    float* pap = misc + 416;    // 128 scalar slots (pap[it])
    float* pa  = misc + 544;    // 512 partials
    float* pb  = misc + 1056;   // 512 partials

    prep<<<1, 256, 0, stream>>>(kern, rw, rks, Kw, Kfw, R, rr, pap);

    // b = conv(img, flipK)
    conv11_wmma<<<384, 256, 0, stream>>>(img, b, Kfw);
    // A x0: t = conv(img,K); Ap = conv(t,flipK); Ap += R (*) img
    conv11_wmma<<<384, 256, 0, stream>>>(img, t, Kw);
    conv11_wmma<<<384, 256, 0, stream>>>(t, Ap, Kfw);
    conv5_add_dot<<<EWB, EWT, 0, stream>>>(img, Ap, R, pa);   // dot discarded
    init_state<<<EWB, EWT, 0, stream>>>(img, b, Ap, x, r, p, pb);
    reduce512<<<1, 256, 0, stream>>>(pb, &rr[0]);

    for (int it = 0; it < NITER; ++it) {
        conv11_wmma<<<384, 256, 0, stream>>>(p, t, Kw);
        conv11_wmma<<<384, 256, 0, stream>>>(t, Ap, Kfw);
        conv5_add_dot<<<EWB, EWT, 0, stream>>>(p, Ap, R, pa);
        reduce512<<<1, 256, 0, stream>>>(pa, &pap[it]);
        axpy_update<<<EWB, EWT, 0, stream>>>(x, r, p, Ap, &rr[it], &pap[it], pb);
        reduce512<<<1, 256, 0, stream>>>(pb, &rr[it + 1]);
        p_update<<<EWB, EWT, 0, stream>>>(p, r, &rr[it + 1], &rr[it]);
    }
}